// GraphConv_Encoder_22428319219806
// MI455X (gfx1250) — compile-verified
//
#include <hip/hip_runtime.h>
#include <hip/hip_bf16.h>
#include <cstdint>
#include <cstddef>

#define DFEAT 128
#define LDSPITCH 132   // float stride; 132 % 64 == 4 -> conflict-free ds_load_b64, 16B-aligned rows

typedef __attribute__((ext_vector_type(2))) float v2f;
typedef __attribute__((ext_vector_type(8))) float v8f;

// ---------------------------------------------------------------------------
// Zero a float buffer (graph-capture-safe replacement for hipMemsetAsync)
// ---------------------------------------------------------------------------
__global__ __launch_bounds__(256) void zero_kernel(float4* __restrict__ p, int n4) {
    int i = blockIdx.x * 256 + threadIdx.x;
    if (i < n4) p[i] = make_float4(0.f, 0.f, 0.f, 0.f);
}

// ---------------------------------------------------------------------------
// Scatter-add: one wave32 per edge, lane l moves float4 at column 4*l.
// agg[dst] += feat[src]   (native f32 global atomics; everything L2-resident)
// edge index layout: ei[0..E-1] = src, ei[E..2E-1] = dst
// ---------------------------------------------------------------------------
__global__ __launch_bounds__(256) void scatter_add_kernel(
    const float* __restrict__ feat,
    const int*   __restrict__ ei,
    float*       __restrict__ agg,
    int E_)
{
    int gid  = blockIdx.x * 256 + threadIdx.x;
    int e    = gid >> 5;
    int lane = gid & 31;
    if (e >= E_) return;

    int src = ei[e];
    int dst = ei[E_ + e];

    const float4 v = *reinterpret_cast<const float4*>(feat + (size_t)src * DFEAT + lane * 4);
    float* p = agg + (size_t)dst * DFEAT + lane * 4;
    unsafeAtomicAdd(p + 0, v.x);
    unsafeAtomicAdd(p + 1, v.y);
    unsafeAtomicAdd(p + 2, v.z);
    unsafeAtomicAdd(p + 3, v.w);
}

// ---------------------------------------------------------------------------
// Fused linear: out = agg @ w_rel + xin @ w_root + bias   (all [*,128] f32)
// One block = 16 node rows; 8 waves, wave w owns output cols [16w, 16w+16).
// Pure-f32 WMMA 16x16x4 (matches f32 reference precision; workload is not
// FLOP-bound so low-precision WMMA buys nothing here).
// ---------------------------------------------------------------------------
__global__ __launch_bounds__(256) void fused_linear_wmma_kernel(
    const float* __restrict__ agg,    // [N,128]  neighbor sums
    const float* __restrict__ xin,    // [N,128]  root features
    const float* __restrict__ w_rel,  // [128,128] row-major [k][n]
    const float* __restrict__ w_root, // [128,128]
    const float* __restrict__ bias,   // [128]
    float*       __restrict__ out)    // [N,128]
{
    __shared__ float sA[16 * LDSPITCH];
    __shared__ float sX[16 * LDSPITCH];

    const int tid  = threadIdx.x;
    const int row0 = blockIdx.x * 16;

    // ---- stage 16x128 tiles of agg and xin into LDS (coalesced b128 loads)
    const float4* gA = reinterpret_cast<const float4*>(agg + (size_t)row0 * DFEAT);
    const float4* gX = reinterpret_cast<const float4*>(xin + (size_t)row0 * DFEAT);
    #pragma unroll
    for (int it = 0; it < 2; ++it) {
        int f   = tid + it * 256;        // 0..511 -> 16 rows x 32 float4
        int r   = f >> 5;
        int c4  = f & 31;
        *reinterpret_cast<float4*>(&sA[r * LDSPITCH + c4 * 4]) = gA[f];
        *reinterpret_cast<float4*>(&sX[r * LDSPITCH + c4 * 4]) = gX[f];
    }
    __syncthreads();

    const int wv   = tid >> 5;          // wave 0..7 -> output col tile
    const int lane = tid & 31;
    const int n0   = wv * 16;
    const int mr   = lane & 15;         // A row / B,C col within tile
    const int kb   = (lane >> 4) << 1;  // 0 for lanes 0-15, 2 for lanes 16-31

    // bias: every C/D VGPR of a lane maps to the same column n0+mr
    v8f acc;
    {
        float bv = bias[n0 + mr];
        #pragma unroll
        for (int j = 0; j < 8; ++j) acc[j] = bv;
    }

    const float* wr = w_rel  + n0 + mr; // B fragment base: W[k][n0+mr]
    const float* wx = w_root + n0 + mr;

    // ---- GEMM 1: acc += agg_tile @ w_rel   (K = 128, 32 wmma steps)
    #pragma unroll 4
    for (int k0 = 0; k0 < DFEAT; k0 += 4) {
        v2f a = *reinterpret_cast<const v2f*>(&sA[mr * LDSPITCH + k0 + kb]);
        v2f b;
        b.x = wr[(size_t)(k0 + kb)     * DFEAT];
        b.y = wr[(size_t)(k0 + kb + 1) * DFEAT];
        acc = __builtin_amdgcn_wmma_f32_16x16x4_f32(false, a, false, b,
                                                    (short)0, acc, false, false);
    }
    // ---- GEMM 2: acc += x_tile @ w_root
    #pragma unroll 4
    for (int k0 = 0; k0 < DFEAT; k0 += 4) {
        v2f a = *reinterpret_cast<const v2f*>(&sX[mr * LDSPITCH + k0 + kb]);
        v2f b;
        b.x = wx[(size_t)(k0 + kb)     * DFEAT];
        b.y = wx[(size_t)(k0 + kb + 1) * DFEAT];
        acc = __builtin_amdgcn_wmma_f32_16x16x4_f32(false, a, false, b,
                                                    (short)0, acc, false, false);
    }

    // ---- store C/D: lane -> col n0+mr; VGPR j -> row j + 8*(lane>=16)
    float* o = out + (size_t)row0 * DFEAT + n0 + mr;
    const int mo = (lane >> 4) * 8;
    #pragma unroll
    for (int j = 0; j < 8; ++j)
        o[(size_t)(mo + j) * DFEAT] = acc[j];
}

// ---------------------------------------------------------------------------
// Launch: zero -> scatter(x) -> linear1 -> h ; zero -> scatter(h) -> linear2
// ---------------------------------------------------------------------------
extern "C" void kernel_launch(void* const* d_in, const int* in_sizes, int n_in,
                              void* d_out, int out_size, void* d_ws, size_t ws_size,
                              hipStream_t stream) {
    const float* x       = (const float*)d_in[0];
    const int*   ei      = (const int*)  d_in[1];
    const float* w_rel1  = (const float*)d_in[2];
    const float* b_rel1  = (const float*)d_in[3];
    const float* w_root1 = (const float*)d_in[4];
    const float* w_rel2  = (const float*)d_in[5];
    const float* b_rel2  = (const float*)d_in[6];
    const float* w_root2 = (const float*)d_in[7];
    float* out = (float*)d_out;

    const int Nn = in_sizes[0] / DFEAT;   // 50000
    const int E_ = in_sizes[1] / 2;       // 800000

    float* agg = (float*)d_ws;                       // N*128 floats
    float* h   = agg + (size_t)Nn * DFEAT;           // N*128 floats

    const int n4        = Nn * DFEAT / 4;
    const int zeroBlks  = (n4 + 255) / 256;
    const int scatBlks  = (E_ + 7) / 8;              // 8 waves (edges) per block
    const int gemmBlks  = Nn / 16;                   // 50000/16 = 3125

    // ---- layer 1
    zero_kernel<<<zeroBlks, 256, 0, stream>>>((float4*)agg, n4);
    scatter_add_kernel<<<scatBlks, 256, 0, stream>>>(x, ei, agg, E_);
    fused_linear_wmma_kernel<<<gemmBlks, 256, 0, stream>>>(
        agg, x, w_rel1, w_root1, b_rel1, h);

    // ---- layer 2
    zero_kernel<<<zeroBlks, 256, 0, stream>>>((float4*)agg, n4);
    scatter_add_kernel<<<scatBlks, 256, 0, stream>>>(h, ei, agg, E_);
    fused_linear_wmma_kernel<<<gemmBlks, 256, 0, stream>>>(
        agg, h, w_rel2, w_root2, b_rel2, out);
}